// HGNN_17729624998271
// MI455X (gfx1250) — compile-verified
//
#include <hip/hip_runtime.h>

// ---------------------------------------------------------------------------
// HGNN graph construction for MI455X (gfx1250, wave32).
// Phase 1: voxel-grid accumulation via global float atomics + atomicMin.
// Phase 2: 6x (8192x8192) ball-query adjacency via V_WMMA_F32_16X16X4_F32:
//          d^2 = |a|^2 - 2 a.b + |b|^2 computed as one K=4 WMMA per 16x16 tile
//          (|a|^2 preloaded into C). Output is ~1.6GB -> HBM-store bound, so
//          the tile epilogue is kept to cmp+cndmask+store with immediate-
//          offset stores; diagonal removal is a wave-uniform rare path.
// ---------------------------------------------------------------------------

typedef __attribute__((ext_vector_type(2))) float v2f;
typedef __attribute__((ext_vector_type(8))) float v8f;

#define NPTS 8192
#define NTILE 512                      // 8192 / 16 tiles per dimension

// voxel grid geometry (EXTENT=30):  dims = ceil(30/vs)+1
__device__ __constant__ float c_vs[3][3]  = {{0.5f,0.5f,1.0f},{0.7f,0.7f,1.2f},{0.9f,0.9f,1.4f}};
__device__ __constant__ int   c_dims[3][3] = {{61,61,31},{44,44,26},{35,35,23}};

static constexpr int NSEG0 = 61*61*31;       // 115351
static constexpr int NSEG1 = 44*44*26;       // 50336
static constexpr int NSEG2 = 35*35*23;       // 28175
__device__ __constant__ int c_nseg[3] = {NSEG0, NSEG1, NSEG2};
static constexpr int ACCB0 = 0;
static constexpr int ACCB1 = ACCB0 + 5*NSEG0;
static constexpr int ACCB2 = ACCB1 + 5*NSEG1;
static constexpr int ACC_END = ACCB2 + 5*NSEG2;          // elems (4B)
__device__ __constant__ int c_accb[3] = {ACCB0, ACCB1, ACCB2};
// packed per-point arrays (4 levels), 16B aligned
static constexpr int WS_ASIDE = ((ACC_END + 3)/4)*4;     // float4 [4*NPTS]
static constexpr int WS_BSIDE = WS_ASIDE + 4*NPTS*4;     // float4 [4*NPTS]
static constexpr int WS_A2    = WS_BSIDE + 4*NPTS*4;     // float  [4*NPTS]

// d_out layout (flattened tuple, element offsets)
static constexpr int    COORD_BASE = 0;                  // [4,8192,3]
static constexpr int    MASK_BASE  = 4*NPTS*3;           // [4,8192]
static constexpr long long NN      = (long long)NPTS*NPTS;
static constexpr long long INTER_BASE = MASK_BASE + 4*NPTS;      // [3,N,N]
static constexpr long long INTRA_BASE = INTER_BASE + 3*NN;       // [3,N,N]

static constexpr float SENT = 1.0e5f;   // masked-point sentinel (A:+, B:-)

// per-adjacency-matrix tables: 0..2 inter (lev i vs i+1), 3..5 intra (lev i+1)
__device__ __constant__ int   c_levA[6] = {0,1,2,1,2,3};
__device__ __constant__ int   c_levB[6] = {1,2,3,1,2,3};
__device__ __constant__ float c_r2[6]   = {1.0f, 2.25f, 4.0f, 2.25f, 4.0f, 6.25f};

__device__ __forceinline__ int voxel_lin(float x, float y, float z, int l) {
    int ix = (int)floorf(x / c_vs[l][0]);
    int iy = (int)floorf(y / c_vs[l][1]);
    int iz = (int)floorf(z / c_vs[l][2]);
    ix = min(max(ix, 0), c_dims[l][0]-1);
    iy = min(max(iy, 0), c_dims[l][1]-1);
    iz = min(max(iz, 0), c_dims[l][2]-1);
    return (ix * c_dims[l][1] + iy) * c_dims[l][2] + iz;
}

// ---- 1) init voxel accumulators: sums/cnt = 0.0f, first = 0xFFFFFFFF -------
__global__ void k_init(unsigned int* __restrict__ ws) {
    int idx = blockIdx.x * blockDim.x + threadIdx.x;
    if (idx >= ACC_END) return;
    unsigned int v = 0u;
    if (idx >= ACCB0 + 4*NSEG0 && idx < ACCB0 + 5*NSEG0) v = 0xFFFFFFFFu;
    if (idx >= ACCB1 + 4*NSEG1 && idx < ACCB1 + 5*NSEG1) v = 0xFFFFFFFFu;
    if (idx >= ACCB2 + 4*NSEG2 && idx < ACCB2 + 5*NSEG2) v = 0xFFFFFFFFu;
    ws[idx] = v;
}

// ---- 2) segment_sum / segment_min via atomics ------------------------------
__global__ void k_accum(const float* __restrict__ pts, unsigned int* __restrict__ ws) {
    int t = blockIdx.x * blockDim.x + threadIdx.x;
    if (t >= 3*NPTS) return;
    int l = t / NPTS, i = t - l*NPTS;
    float x = pts[i*4+0], y = pts[i*4+1], z = pts[i*4+2];
    int lin = voxel_lin(x, y, z, l);
    float* f = (float*)ws;
    int b = c_accb[l], ns = c_nseg[l];
    atomicAdd(&f[b + 0*ns + lin], x);
    atomicAdd(&f[b + 1*ns + lin], y);
    atomicAdd(&f[b + 2*ns + lin], z);
    atomicAdd(&f[b + 3*ns + lin], 1.0f);
    atomicMin(&ws[b + 4*ns + lin], (unsigned int)i);
}

// ---- 3) per-point coords/masks + packed WMMA operands ----------------------
__global__ void k_prep(const float* __restrict__ pts, unsigned int* __restrict__ ws,
                       float* __restrict__ out) {
    int t = blockIdx.x * blockDim.x + threadIdx.x;
    if (t >= 4*NPTS) return;
    int lev = t >> 13, i = t & (NPTS-1);
    float x = pts[i*4+0], y = pts[i*4+1], z = pts[i*4+2];
    float* f = (float*)ws;
    float cx, cy, cz; int mask;
    if (lev == 0) {
        cx = x; cy = y; cz = z; mask = 1;
    } else {
        int l = lev - 1;
        int lin = voxel_lin(x, y, z, l);
        int b = c_accb[l], ns = c_nseg[l];
        float inv = 1.0f / fmaxf(f[b + 3*ns + lin], 1.0f);
        cx = f[b + 0*ns + lin] * inv;
        cy = f[b + 1*ns + lin] * inv;
        cz = f[b + 2*ns + lin] * inv;
        mask = (ws[b + 4*ns + lin] == (unsigned int)i) ? 1 : 0;
    }
    // coords_out (zeroed where mask false) + masks_out (as 0/1 float)
    int pi = lev*NPTS + i;
    out[COORD_BASE + pi*3 + 0] = mask ? cx : 0.0f;
    out[COORD_BASE + pi*3 + 1] = mask ? cy : 0.0f;
    out[COORD_BASE + pi*3 + 2] = mask ? cz : 0.0f;
    out[MASK_BASE + pi] = mask ? 1.0f : 0.0f;
    // packed A side: (-2ax,-2ay,-2az, 1); sentinel +SENT if masked out
    float ax = mask ? cx :  SENT, ay = mask ? cy :  SENT, az = mask ? cz :  SENT;
    float bx = mask ? cx : -SENT, by = mask ? cy : -SENT, bz = mask ? cz : -SENT;
    float4* AS = (float4*)(f + WS_ASIDE);
    float4* BS = (float4*)(f + WS_BSIDE);
    AS[pi] = make_float4(-2.0f*ax, -2.0f*ay, -2.0f*az, 1.0f);
    BS[pi] = make_float4(bx, by, bz, bx*bx + by*by + bz*bz);
    f[WS_A2 + pi] = ax*ax + ay*ay + az*az;
}

// ---- 4) adjacency: one 16x16 tile per wave via V_WMMA_F32_16X16X4_F32 ------
__global__ void k_adj(const unsigned int* __restrict__ ws, float* __restrict__ out) {
    const float*  f  = (const float*)ws;
    const float4* AS = (const float4*)(f + WS_ASIDE);
    const float4* BS = (const float4*)(f + WS_BSIDE);
    const float*  A2 = f + WS_A2;

    int wid  = blockIdx.x * 8 + (threadIdx.x >> 5);            // wave32
    int lane = threadIdx.x & 31;
    int hi   = lane >> 4;        // 0: lanes 0-15, 1: lanes 16-31
    int l16  = lane & 15;

    // all powers of two -> shifts/masks
    int mat = wid >> 18;                 // / (512*512)
    int tt  = wid & (NTILE*NTILE - 1);
    int tm  = tt >> 9;
    int tn  = tt & (NTILE - 1);

    int la = c_levA[mat], lb = c_levB[mat];
    // A 16x4 f32 layout: lanes 0-15 -> M=lane,{K0,K1}; lanes 16-31 -> M,{K2,K3}
    float4 av = AS[la*NPTS + tm*16 + l16];
    float4 bv = BS[lb*NPTS + tn*16 + l16];
    v2f A, B;
    A[0] = hi ? av.z : av.x;  A[1] = hi ? av.w : av.y;
    B[0] = hi ? bv.z : bv.x;  B[1] = hi ? bv.w : bv.y;
    // C preloaded with |a|^2 per output row: VGPR v -> M = v + (hi?8:0).
    // Region is 32B-aligned (mbase multiple of 8) -> two b128 loads.
    int mbase = tm*16 + (hi ? 8 : 0);
    const float4* a24 = (const float4*)(A2 + la*NPTS + mbase);
    float4 ca = a24[0], cb = a24[1];
    v8f C = {ca.x, ca.y, ca.z, ca.w, cb.x, cb.y, cb.z, cb.w};

    // D = A x B + C  ==  d^2 for the 16x16 tile
    v8f D = __builtin_amdgcn_wmma_f32_16x16x4_f32(
        /*neg_a=*/false, A, /*neg_b=*/false, B,
        /*c_mod=*/(short)0, C, /*reuse_a=*/false, /*reuse_b=*/false);

    float r2 = c_r2[mat];
    long long obase = (mat >= 3) ? (INTRA_BASE + (long long)(mat-3)*NN)
                                 : (INTER_BASE + (long long)mat*NN);
    int ncol = tn*16 + l16;
    // single per-lane base pointer; v-th row store folds to immediate offset
    float* p = out + (obase + (long long)mbase * NPTS + ncol);

    if ((mat >= 3) & (tm == tn)) {
        // rare wave-uniform path: tile contains diagonal elements (& ~eye)
#pragma unroll
        for (int v = 0; v < 8; ++v) {
            float val = (D[v] <= r2) ? 1.0f : 0.0f;
            if (mbase + v == ncol) val = 0.0f;
            p[v * NPTS] = val;
        }
    } else {
        // hot path: 2 VALU + 1 store per element
#pragma unroll
        for (int v = 0; v < 8; ++v)
            p[v * NPTS] = (D[v] <= r2) ? 1.0f : 0.0f;
    }
}

extern "C" void kernel_launch(void* const* d_in, const int* in_sizes, int n_in,
                              void* d_out, int out_size, void* d_ws, size_t ws_size,
                              hipStream_t stream) {
    const float* pts = (const float*)d_in[0];
    float* out = (float*)d_out;
    unsigned int* ws = (unsigned int*)d_ws;

    (void)in_sizes; (void)n_in; (void)out_size; (void)ws_size;

    // 1) zero/initialize voxel accumulators (~970K elems)
    k_init<<<dim3((ACC_END + 255)/256), dim3(256), 0, stream>>>(ws);
    // 2) voxel accumulation: 3 levels x 8192 points
    k_accum<<<dim3((3*NPTS + 255)/256), dim3(256), 0, stream>>>(pts, ws);
    // 3) coords/masks + packed WMMA operands: 4 levels x 8192 points
    k_prep<<<dim3((4*NPTS + 255)/256), dim3(256), 0, stream>>>(pts, ws, out);
    // 4) 6 matrices x 512x512 tiles, 8 waves (tiles) per 256-thread block
    const int total_tiles = 6 * NTILE * NTILE;          // 1,572,864
    k_adj<<<dim3(total_tiles / 8), dim3(256), 0, stream>>>(ws, out);
}